// Net_57269093925097
// MI455X (gfx1250) — compile-verified
//
#include <hip/hip_runtime.h>
#include <stdint.h>

// ---------------------------------------------------------------------------
// Bulyan/Krum selection for x:[1, D=500000, N=50] fp32.
// Output == column i* of x where i* = Krum argmin.
// Pipeline: (1) G = X^T X via V_WMMA_F32_16X16X4_F32 (fp32 matrix unit),
//               staged into LDS via GLOBAL_LOAD_ASYNC_TO_LDS_B32 when available,
//           (2) deterministic partial-sum reduction,
//           (3) distance/score/argmin (one block),
//           (4) gather winning column (L2-resident second pass).
// ---------------------------------------------------------------------------

typedef __attribute__((ext_vector_type(2))) float v2f;
typedef __attribute__((ext_vector_type(8))) float v8f;

#define NCLI 50          // clients (N)
#define PADN 64          // padded to 4x4 tiles of 16x16
#define LSTRIDE 72       // LDS row stride in floats (2*72 mod 64 = 16 -> half-waves hit disjoint bank windows)
#define ROWS 64          // K rows staged per LDS chunk
#define GRAM_THREADS 512 // 16 waves -> one 16x16 tile per wave

#if __has_builtin(__builtin_amdgcn_global_load_async_to_lds_b32)
#define HAVE_ASYNC_LDS 1
#endif

#ifdef HAVE_ASYNC_LDS
typedef __attribute__((address_space(1))) int* as1_i32p;
typedef __attribute__((address_space(3))) int* as3_i32p;

__device__ __forceinline__ void async_copy_b32(const float* g, float* l) {
    int* gi = (int*)(uintptr_t)g;          // generic, const dropped
    int* li = (int*)(uintptr_t)l;
    __builtin_amdgcn_global_load_async_to_lds_b32(
        (as1_i32p)gi, (as3_i32p)li, /*offset=*/0, /*cpol=*/0);
}
__device__ __forceinline__ void wait_async() {
#if __has_builtin(__builtin_amdgcn_s_wait_asynccnt)
    __builtin_amdgcn_s_wait_asynccnt(0);
#else
    asm volatile("s_wait_asynccnt 0" ::: "memory");
#endif
}
#endif

// ---------------- Kernel 1: per-block partial Gram via fp32 WMMA -----------
__global__ void __launch_bounds__(GRAM_THREADS)
gram_kernel(const float* __restrict__ X, float* __restrict__ partials,
            int D, int nblk) {
    __shared__ float lds[ROWS * LSTRIDE];

    const int tid  = threadIdx.x;
    const int lane = tid & 31;
    const int wave = tid >> 5;          // 0..15
    const int tm   = wave >> 2;         // tile row  0..3
    const int tn   = wave & 3;          // tile col  0..3
    const int lhalf = lane >> 4;        // half-wave
    const int lm    = lane & 15;

    // chunk of rows for this block, 64-aligned
    int chunk = ((D + nblk - 1) / nblk + (ROWS - 1)) & ~(ROWS - 1);
    int d0 = blockIdx.x * chunk;
    int d1 = d0 + chunk; if (d1 > D) d1 = D;

    // zero the padded columns 50..63 once (they are never overwritten)
    for (int t = tid; t < ROWS * (PADN - NCLI); t += GRAM_THREADS) {
        int r = t / (PADN - NCLI);
        int c = NCLI + t - r * (PADN - NCLI);
        lds[r * LSTRIDE + c] = 0.0f;
    }

    v8f acc = {};                        // 16x16 fp32 accumulator tile

    const int acol = tm * 16 + lm;       // A-fragment column (M)
    const int bcol = tn * 16 + lm;       // B-fragment column (N)

    for (int d = d0; d < d1; d += ROWS) {
        const int rows  = (d1 - d < ROWS) ? (d1 - d) : ROWS;
        const int total = rows * NCLI;
        const float* __restrict__ src = X + (size_t)d * NCLI;   // contiguous chunk

        __syncthreads();                 // previous chunk fully consumed

        // stage ROWS x 50 floats (flat-contiguous in global) into LDS
        for (int t = tid; t < ROWS * NCLI; t += GRAM_THREADS) {
            int r = t / NCLI;
            int c = t - r * NCLI;
#ifdef HAVE_ASYNC_LDS
            if (t < total) {
                async_copy_b32(src + t, &lds[r * LSTRIDE + c]); // mem -> LDS, no VGPR round-trip
            } else {
                lds[r * LSTRIDE + c] = 0.0f;                    // tail zero-pad (disjoint addrs)
            }
#else
            lds[r * LSTRIDE + c] = (t < total) ? src[t] : 0.0f;
#endif
        }
        // bounded speculative prefetch: threads 0..99 each grab one 128B line
        // covering exactly the next 3200-float chunk, only if fully in range.
        if (tid < 100 && (d + 2 * ROWS) <= d1) {
            __builtin_prefetch(src + ROWS * NCLI + tid * 32, 0, 0);
        }
#ifdef HAVE_ASYNC_LDS
        wait_async();                    // ASYNCcnt == 0 before cross-wave barrier
#endif
        __syncthreads();

        // 16 K-steps of 4: D += A(16x4) * B(4x16)
        #pragma unroll 4
        for (int kb = 0; kb < ROWS; kb += 4) {
            const int r0 = kb + 2 * lhalf;  // fp32 A/B layout: K = 2*(lane>=16)+component
            v2f a, b;
            a.x = lds[(r0 + 0) * LSTRIDE + acol];
            a.y = lds[(r0 + 1) * LSTRIDE + acol];
            b.x = lds[(r0 + 0) * LSTRIDE + bcol];
            b.y = lds[(r0 + 1) * LSTRIDE + bcol];
            acc = __builtin_amdgcn_wmma_f32_16x16x4_f32(
                false, a, false, b, (short)0, acc, false, false);
        }
    }

    // write this block's 64x64 partial Gram (each element owned by one lane)
    float* __restrict__ P = partials + (size_t)blockIdx.x * (PADN * PADN);
    const int i0 = tm * 16 + 8 * lhalf;          // C/D layout: M = v + 8*lhalf
    #pragma unroll
    for (int v = 0; v < 8; ++v) {
        P[(i0 + v) * PADN + bcol] = acc[v];
    }
}

// ---------------- Kernel 2: deterministic reduction of partials ------------
__global__ void reduce_kernel(const float* __restrict__ partials,
                              float* __restrict__ G, int nblk) {
    int e = blockIdx.x * blockDim.x + threadIdx.x;   // 0..4095
    float s = 0.0f;
    for (int b = 0; b < nblk; ++b) s += partials[(size_t)b * (PADN * PADN) + e];
    G[e] = s;
}

// ---------------- Kernel 3: distances, Krum scores, argmin -----------------
__global__ void score_kernel(const float* __restrict__ G, int* __restrict__ istar) {
    __shared__ float sdist[NCLI][NCLI + 2];
    __shared__ float sscore[64];

    const int i = threadIdx.x;
    float score = 3.402823466e+38f;
    if (i < NCLI) {
        const float gii = G[i * PADN + i];
        float tot = 0.0f;
        for (int j = 0; j < NCLI; ++j) {
            float d2 = gii + G[j * PADN + j] - 2.0f * G[i * PADN + j];
            float dd = (d2 > 0.0f) ? sqrtf(d2) : 0.0f;
            sdist[i][j] = dd;
            tot += dd;
        }
        // score = sum of (k+1)=39 smallest = total - 11 largest  (n=50,f=10,k=38)
        for (int r = 0; r < NCLI - (NCLI - 10 - 2 + 1); ++r) {   // 11 iterations
            float m = -1.0f; int mi = 0;
            for (int j = 0; j < NCLI; ++j) {
                float v = sdist[i][j];
                if (v > m) { m = v; mi = j; }
            }
            tot -= m;
            sdist[i][mi] = -1.0f;
        }
        score = tot;
    }
    sscore[i] = score;
    __syncthreads();
    if (i == 0) {
        float best = sscore[0]; int bi = 0;
        for (int j = 1; j < NCLI; ++j) {
            if (sscore[j] < best) { best = sscore[j]; bi = j; }  // first-index ties
        }
        *istar = bi;
    }
}

// ---------------- Kernel 4: gather winning column --------------------------
__global__ void gather_kernel(const float* __restrict__ X,
                              const int* __restrict__ istar,
                              float* __restrict__ out, int D) {
    const int is = *istar;
    for (int d = blockIdx.x * blockDim.x + threadIdx.x; d < D;
         d += gridDim.x * blockDim.x) {
        out[d] = X[(size_t)d * NCLI + is];
    }
}

// ---------------------------------------------------------------------------
extern "C" void kernel_launch(void* const* d_in, const int* in_sizes, int n_in,
                              void* d_out, int out_size, void* d_ws, size_t ws_size,
                              hipStream_t stream) {
    const float* X = (const float*)d_in[0];
    const int total = in_sizes[0];       // 25,000,000
    const int D = total / NCLI;          // 500,000

    // workspace: [nblk * 64*64] partials | [64*64] G | [1] istar
    float* partials = (float*)d_ws;
    size_t elems = ws_size / sizeof(float);
    int nblk = 256;
    if (elems > (PADN * PADN + 16)) {
        int maxblk = (int)((elems - (PADN * PADN + 16)) / (PADN * PADN));
        if (nblk > maxblk) nblk = maxblk;
    }
    if (nblk < 1) nblk = 1;
    float* G = partials + (size_t)nblk * (PADN * PADN);
    int* istar = (int*)(G + PADN * PADN);

    gram_kernel  <<<nblk, GRAM_THREADS, 0, stream>>>(X, partials, D, nblk);
    reduce_kernel<<<(PADN * PADN) / 256, 256, 0, stream>>>(partials, G, nblk);
    score_kernel <<<1, 64, 0, stream>>>(G, istar);
    gather_kernel<<<1024, 256, 0, stream>>>(X, istar, (float*)d_out, D);
}